// AspectMatcher_33827162423826
// MI455X (gfx1250) — compile-verified
//
#include <hip/hip_runtime.h>
#include <hip/hip_bf16.h>
#include <math.h>

typedef float v2f __attribute__((ext_vector_type(2)));
typedef float v8f __attribute__((ext_vector_type(8)));

// -----------------------------------------------------------------------------
// Generic NT GEMM:  C[M,N] = A[M,K] * B[N,K]^T (+ bias[n] if bias != nullptr)
// A row-major (lda = K), Bn row-major [N,K] (ldb = K), C row-major (ldc = N).
// Wave tile: 32x64 (2 M-tiles x 4 N-tiles of 16x16). Block: 8 waves -> 64x256.
// Uses V_WMMA_F32_16X16X4_F32 (exact fp32, matches reference precision).
// Fragment layouts per CDNA5 ISA 7.12.2:
//   A(16x4):  element (m,k) in lane = m + 16*(k>=2), vgpr = k&1  -> float2/lane
//   B(4x16):  element (k,n) in lane = n + 16*(k>=2), vgpr = k&1  -> float2/lane
//   C(16x16): vgpr i holds (m = i + 8*(lane>=16), n = lane&15)
// -----------------------------------------------------------------------------
__global__ __launch_bounds__(256) void wmma_gemm_nt_f32(
    const float* __restrict__ A, const float* __restrict__ Bn,
    const float* __restrict__ bias, float* __restrict__ C,
    int K, int N,
    long long strideA, long long strideB, long long strideC)
{
  const int bz = blockIdx.z;
  A  += (long long)bz * strideA;
  Bn += (long long)bz * strideB;
  C  += (long long)bz * strideC;

  const int lane = threadIdx.x & 31;
  const int wave = threadIdx.x >> 5;
  const int wm   = wave & 1;          // 0..1 : M direction within block
  const int wn   = wave >> 1;         // 0..3 : N direction within block
  const int m0   = blockIdx.y * 64  + wm * 32;
  const int n0   = blockIdx.x * 256 + wn * 64;

  const int lr = lane & 15;           // tile-row (A) / tile-col (B) selector
  const int kh = (lane >> 4) << 1;    // 0 or 2 : which K-subpair this lane holds

  v8f acc[2][4];
#pragma unroll
  for (int i = 0; i < 2; ++i)
#pragma unroll
    for (int j = 0; j < 4; ++j)
      acc[i][j] = (v8f){0.f, 0.f, 0.f, 0.f, 0.f, 0.f, 0.f, 0.f};

  const float* a0p = A  + (long long)(m0 +      lr) * K + kh;
  const float* a1p = A  + (long long)(m0 + 16 + lr) * K + kh;
  const float* b0p = Bn + (long long)(n0 +      lr) * K + kh;
  const float* b1p = Bn + (long long)(n0 + 16 + lr) * K + kh;
  const float* b2p = Bn + (long long)(n0 + 32 + lr) * K + kh;
  const float* b3p = Bn + (long long)(n0 + 48 + lr) * K + kh;

#pragma unroll 2
  for (int kk = 0; kk < K; kk += 4) {
    v2f af[2], bf[4];
    af[0] = *(const v2f*)(a0p + kk);
    af[1] = *(const v2f*)(a1p + kk);
    bf[0] = *(const v2f*)(b0p + kk);
    bf[1] = *(const v2f*)(b1p + kk);
    bf[2] = *(const v2f*)(b2p + kk);
    bf[3] = *(const v2f*)(b3p + kk);
#pragma unroll
    for (int i = 0; i < 2; ++i)
#pragma unroll
      for (int j = 0; j < 4; ++j)
        acc[i][j] = __builtin_amdgcn_wmma_f32_16x16x4_f32(
            /*neg_a=*/false, af[i], /*neg_b=*/false, bf[j],
            /*c_mod=*/(short)0, acc[i][j],
            /*reuse_a=*/false, /*reuse_b=*/false);
  }

  // Store (with optional bias). Column for this lane within N-tile j:
  const int mh = (lane >> 4) << 3;    // +8 row offset for upper half-wave
#pragma unroll
  for (int j = 0; j < 4; ++j) {
    const int col = n0 + j * 16 + lr;
    const float bv = bias ? bias[col] : 0.f;
#pragma unroll
    for (int i = 0; i < 2; ++i) {
      const long long rowbase = (long long)(m0 + i * 16 + mh) * N + col;
#pragma unroll
      for (int r = 0; r < 8; ++r) {
        C[rowbase + (long long)r * N] = acc[i][j][r] + bv;
      }
    }
  }
}

// -----------------------------------------------------------------------------
// Per-row log-softmax + masked CE accumulation.
// One 256-thread block per (b,s) row of logits[b,s,:]  (S = 1024).
// acc[0] += -logp(tgt)  for masked rows;  acc[1] += 1.0 per masked row.
// -----------------------------------------------------------------------------
__global__ __launch_bounds__(256) void row_ce_kernel(
    const float* __restrict__ logits, const int* __restrict__ mask,
    const int* __restrict__ tgt, float* __restrict__ acc, int S)
{
  const int r = blockIdx.x;
  const float* row = logits + (long long)r * S;
  __shared__ float red[256];

  float lmax = -INFINITY;
  for (int t = threadIdx.x; t < S; t += 256) lmax = fmaxf(lmax, row[t]);
  red[threadIdx.x] = lmax;
  __syncthreads();
#pragma unroll
  for (int s2 = 128; s2 > 0; s2 >>= 1) {
    if (threadIdx.x < s2)
      red[threadIdx.x] = fmaxf(red[threadIdx.x], red[threadIdx.x + s2]);
    __syncthreads();
  }
  const float m = red[0];
  __syncthreads();

  float lsum = 0.f;
  for (int t = threadIdx.x; t < S; t += 256) lsum += expf(row[t] - m);
  red[threadIdx.x] = lsum;
  __syncthreads();
#pragma unroll
  for (int s2 = 128; s2 > 0; s2 >>= 1) {
    if (threadIdx.x < s2)
      red[threadIdx.x] += red[threadIdx.x + s2];
    __syncthreads();
  }

  if (threadIdx.x == 0 && mask[r] != 0) {
    const float lp = row[tgt[r]] - m - logf(red[0]);
    atomicAdd(&acc[0], -lp);
    atomicAdd(&acc[1], 1.0f);
  }
}

__global__ void zero_acc_kernel(float* acc) {
  if (threadIdx.x < 2) acc[threadIdx.x] = 0.f;
}

__global__ void finalize_kernel(const float* __restrict__ acc,
                                float* __restrict__ out) {
  if (threadIdx.x == 0) out[0] = acc[0] / fmaxf(acc[1], 1.0f);
}

// -----------------------------------------------------------------------------
extern "C" void kernel_launch(void* const* d_in, const int* in_sizes, int n_in,
                              void* d_out, int out_size, void* d_ws, size_t ws_size,
                              hipStream_t stream) {
  constexpr int B = 32, S = 1024, H = 1024;

  const float* X    = (const float*)d_in[0];  // [B,S,H]
  const int*   mask = (const int*)d_in[1];    // [B,S]
  const int*   tgt  = (const int*)d_in[2];    // [B,S]
  const float* W    = (const float*)d_in[3];  // [H,H]
  const float* bias = (const float*)d_in[4];  // [H]

  float* out    = (float*)d_out;              // [0]=loss, [1..] = logits [B,S,S]
  float* logits = out + 1;

  float* accws = (float*)d_ws;                        // 2 floats: sum, count
  float* Q     = (float*)((char*)d_ws + 256);         // [B*S, H] = 134 MB

  dim3 blk(256);

  // GEMM 1: Q = X @ W^T + b       (M = B*S = 32768, N = H, K = H)
  {
    dim3 grid(H / 256, (B * S) / 64, 1);
    wmma_gemm_nt_f32<<<grid, blk, 0, stream>>>(X, W, bias, Q, H, H, 0, 0, 0);
  }

  // GEMM 2: logits[b] = Q[b] @ X[b]^T   (per-batch M = N = S, K = H)
  {
    dim3 grid(S / 256, S / 64, B);
    wmma_gemm_nt_f32<<<grid, blk, 0, stream>>>(
        Q, X, nullptr, logits, H, S,
        (long long)S * H, (long long)S * H, (long long)S * S);
  }

  // Masked cross-entropy over rows.
  zero_acc_kernel<<<1, 32, 0, stream>>>(accws);
  row_ce_kernel<<<B * S, 256, 0, stream>>>(logits, mask, tgt, accws, S);
  finalize_kernel<<<1, 32, 0, stream>>>(accws, out);
}